// GATConvManual_67095979098991
// MI455X (gfx1250) — compile-verified
//
#include <hip/hip_runtime.h>
#include <hip/hip_bf16.h>
#include <math.h>

typedef __attribute__((ext_vector_type(2))) float v2f;
typedef __attribute__((ext_vector_type(8))) float v8f;

#define N_NODES   50000
#define N_EDGES   800000
#define N_ENTRIES (N_EDGES + N_NODES)   // edges + self loops
#define IN_CH     128
#define HEADS     4
#define OUT_CH    32
#define HC        128                    // HEADS*OUT_CH

__device__ __forceinline__ float lrelu(float v) { return v > 0.0f ? v : 0.2f * v; }

__device__ __forceinline__ void edge_sd(int i, const int* __restrict__ ei, int& src, int& dst) {
    if (i < N_EDGES) { src = ei[i]; dst = ei[N_EDGES + i]; }
    else             { src = i - N_EDGES; dst = src; }      // self loop
}

// ---------------------------------------------------------------------------
// 1) init: out = bias broadcast, node_max = -inf, node_sum = 0
// ---------------------------------------------------------------------------
__global__ void gat_init(float* __restrict__ out, const float* __restrict__ bias,
                         float* __restrict__ nmax, float* __restrict__ nsum) {
    int idx = blockIdx.x * 256 + threadIdx.x;           // covers N_NODES*HC exactly
    out[idx] = bias[idx & (HC - 1)];
    if (idx < N_NODES * HEADS) { nmax[idx] = -3.0e38f; nsum[idx] = 0.0f; }
}

// ---------------------------------------------------------------------------
// 2) h = x @ W^T  via V_WMMA_F32_16X16X4_F32.
//    Block = 256 threads = 8 waves. Block computes a 16-row x 128-col tile.
//    Wave w computes columns [16w, 16w+16). K loop: 4 chunks of 32, each
//    chunk stages W^T[32][128] in LDS, 8 WMMA(K=4) steps per chunk.
// ---------------------------------------------------------------------------
__global__ __launch_bounds__(256) void gat_gemm(const float* __restrict__ x,
                                                const float* __restrict__ W,
                                                float* __restrict__ h) {
    __shared__ float xs[16 * 132];        // x tile, padded stride 132 (conflict-free)
    __shared__ float wt[32 * 132];        // W^T chunk: wt[k'][n]

    const int t    = threadIdx.x;
    const int wave = t >> 5;
    const int lane = t & 31;
    const int r0   = blockIdx.x * 16;     // 50000 = 3125 * 16, exact

    // ---- stage x tile [16][128] (coalesced float4) ----
    {
        int r = t >> 5;                   // 0..7
        int c = (t & 31) * 4;             // 0..124
        #pragma unroll
        for (int i = 0; i < 2; ++i) {
            int rr = r + 8 * i;
            float4 v = *(const float4*)(x + (size_t)(r0 + rr) * IN_CH + c);
            xs[rr * 132 + c + 0] = v.x;
            xs[rr * 132 + c + 1] = v.y;
            xs[rr * 132 + c + 2] = v.z;
            xs[rr * 132 + c + 3] = v.w;
        }
    }

    const int row  = lane & 15;           // A-matrix M / B-matrix N offset
    const int kh   = (lane >> 4) * 2;     // K sub-offset per lane half (0 or 2)
    const int n0   = wave * 16;
    const int ncol = n0 + row;

    v8f acc = {0.f, 0.f, 0.f, 0.f, 0.f, 0.f, 0.f, 0.f};

    for (int kc = 0; kc < IN_CH; kc += 32) {
        __syncthreads();
        // ---- stage W^T chunk: wt[k'][n] = W[n][kc+k'], coalesced float4 ----
        #pragma unroll
        for (int i = 0; i < 4; ++i) {
            int q    = t + 256 * i;       // 1024 float4 loads total
            int flat = q * 4;             // over (n, k') with k' fastest
            int n    = flat >> 5;         // 0..127
            int k4   = flat & 31;         // 0,4,...,28
            float4 v = *(const float4*)(W + (size_t)n * IN_CH + kc + k4);
            wt[(k4 + 0) * 132 + n] = v.x;
            wt[(k4 + 1) * 132 + n] = v.y;
            wt[(k4 + 2) * 132 + n] = v.z;
            wt[(k4 + 3) * 132 + n] = v.w;
        }
        __syncthreads();

        #pragma unroll
        for (int kk = 0; kk < 32; kk += 4) {
            v2f a, b;
            // A (16x4 f32): lanes 0-15 hold K={k0,k0+1}, lanes 16-31 K={k0+2,k0+3}
            a.x = xs[row * 132 + kc + kk + kh + 0];
            a.y = xs[row * 132 + kc + kk + kh + 1];
            // B (4x16 f32): B[k][n] = W[n][k], mirrored lane-half K split
            b.x = wt[(kk + kh + 0) * 132 + ncol];
            b.y = wt[(kk + kh + 1) * 132 + ncol];
            acc = __builtin_amdgcn_wmma_f32_16x16x4_f32(
                false, a, false, b, (short)0, acc, false, false);
        }
    }

    // ---- store D: VGPR r -> rows {r, r+8} by lane half, col = n0 + (lane&15)
    const int mrow = r0 + ((lane >> 4) * 8);
    const int col  = n0 + (lane & 15);
    #pragma unroll
    for (int r = 0; r < 8; ++r) {
        h[(size_t)(mrow + r) * HC + col] = acc[r];
    }
}

// ---------------------------------------------------------------------------
// 3) per-node scores: s_src[n,h] = <h[n,h,:], att[h,:C]>, s_dst with att[h,C:]
//    One wave per node; lane covers 4 channels; 8-lane shuffle reduce per head.
// ---------------------------------------------------------------------------
__global__ void gat_scores(const float* __restrict__ h, const float* __restrict__ att,
                           float* __restrict__ ssrc, float* __restrict__ sdst) {
    int node = blockIdx.x * 8 + (threadIdx.x >> 5);   // 50000 = 6250*8, exact
    int lane = threadIdx.x & 31;
    int c    = lane * 4;
    int head = lane >> 3;
    int cc   = c & 31;

    float4 hv = *(const float4*)(h + (size_t)node * HC + c);
    float4 as = *(const float4*)(att + head * (2 * OUT_CH) + cc);
    float4 ad = *(const float4*)(att + head * (2 * OUT_CH) + OUT_CH + cc);

    float ps = hv.x * as.x + hv.y * as.y + hv.z * as.z + hv.w * as.w;
    float pd = hv.x * ad.x + hv.y * ad.y + hv.z * ad.z + hv.w * ad.w;

    #pragma unroll
    for (int off = 4; off >= 1; off >>= 1) {
        ps += __shfl_xor(ps, off, 8);
        pd += __shfl_xor(pd, off, 8);
    }
    if ((lane & 7) == 0) {
        ssrc[node * HEADS + head] = ps;
        sdst[node * HEADS + head] = pd;
    }
}

// ---------------------------------------------------------------------------
// 4) segment max over incoming edges (float atomic max -> GLOBAL_ATOMIC_MAX_NUM_F32)
// ---------------------------------------------------------------------------
__global__ void gat_edge_max(const int* __restrict__ ei,
                             const float* __restrict__ ssrc, const float* __restrict__ sdst,
                             float* __restrict__ nmax) {
    int i = blockIdx.x * 256 + threadIdx.x;
    if (i >= N_ENTRIES) return;
    int src, dst; edge_sd(i, ei, src, dst);
    float4 a = *(const float4*)(ssrc + src * HEADS);
    float4 b = *(const float4*)(sdst + dst * HEADS);
    atomicMax(&nmax[dst * HEADS + 0], lrelu(a.x + b.x));
    atomicMax(&nmax[dst * HEADS + 1], lrelu(a.y + b.y));
    atomicMax(&nmax[dst * HEADS + 2], lrelu(a.z + b.z));
    atomicMax(&nmax[dst * HEADS + 3], lrelu(a.w + b.w));
}

// ---------------------------------------------------------------------------
// 5) segment sum of alpha = exp(e - max[dst]) (recompute e; cheaper than storing)
// ---------------------------------------------------------------------------
__global__ void gat_edge_sum(const int* __restrict__ ei,
                             const float* __restrict__ ssrc, const float* __restrict__ sdst,
                             const float* __restrict__ nmax, float* __restrict__ nsum) {
    int i = blockIdx.x * 256 + threadIdx.x;
    if (i >= N_ENTRIES) return;
    int src, dst; edge_sd(i, ei, src, dst);
    float4 a  = *(const float4*)(ssrc + src * HEADS);
    float4 b  = *(const float4*)(sdst + dst * HEADS);
    float4 mx = *(const float4*)(nmax + dst * HEADS);
    atomicAdd(&nsum[dst * HEADS + 0], __expf(lrelu(a.x + b.x) - mx.x));
    atomicAdd(&nsum[dst * HEADS + 1], __expf(lrelu(a.y + b.y) - mx.y));
    atomicAdd(&nsum[dst * HEADS + 2], __expf(lrelu(a.z + b.z) - mx.z));
    atomicAdd(&nsum[dst * HEADS + 3], __expf(lrelu(a.w + b.w) - mx.w));
}

// ---------------------------------------------------------------------------
// 6) aggregate: out[dst] += coeff[head] * h[src]. One wave per edge entry;
//    lane handles 4 channels (float4 gather + 4 f32 atomic adds).
// ---------------------------------------------------------------------------
__global__ void gat_aggregate(const int* __restrict__ ei,
                              const float* __restrict__ ssrc, const float* __restrict__ sdst,
                              const float* __restrict__ nmax, const float* __restrict__ nsum,
                              const float* __restrict__ h, float* __restrict__ out) {
    int entry = blockIdx.x * 8 + (threadIdx.x >> 5);  // 850000 = 106250*8, exact
    int lane  = threadIdx.x & 31;
    int src, dst; edge_sd(entry, ei, src, dst);

    float4 a  = *(const float4*)(ssrc + src * HEADS);
    float4 b  = *(const float4*)(sdst + dst * HEADS);
    float4 mx = *(const float4*)(nmax + dst * HEADS);
    float4 sm = *(const float4*)(nsum + dst * HEADS);

    float c0 = __expf(lrelu(a.x + b.x) - mx.x) / fmaxf(sm.x, 1e-10f);
    float c1 = __expf(lrelu(a.y + b.y) - mx.y) / fmaxf(sm.y, 1e-10f);
    float c2 = __expf(lrelu(a.z + b.z) - mx.z) / fmaxf(sm.z, 1e-10f);
    float c3 = __expf(lrelu(a.w + b.w) - mx.w) / fmaxf(sm.w, 1e-10f);

    int c    = lane * 4;
    int head = lane >> 3;
    float w  = (head < 2) ? (head == 0 ? c0 : c1) : (head == 2 ? c2 : c3);

    float4 hv = *(const float4*)(h + (size_t)src * HC + c);
    float* o  = out + (size_t)dst * HC + c;
    atomicAdd(o + 0, w * hv.x);
    atomicAdd(o + 1, w * hv.y);
    atomicAdd(o + 2, w * hv.z);
    atomicAdd(o + 3, w * hv.w);
}

// ---------------------------------------------------------------------------
extern "C" void kernel_launch(void* const* d_in, const int* in_sizes, int n_in,
                              void* d_out, int out_size, void* d_ws, size_t ws_size,
                              hipStream_t stream) {
    const float* x    = (const float*)d_in[0];
    const int*   ei   = (const int*)  d_in[1];   // edge_index flat [2, E]
    const float* W    = (const float*)d_in[2];   // [HC, IN_CH]
    const float* att  = (const float*)d_in[3];   // [HEADS, 2*OUT_CH]
    const float* bias = (const float*)d_in[4];   // [HC]
    float* out = (float*)d_out;

    char* ws = (char*)d_ws;
    float* h    = (float*)(ws);                          // N*HC f32 = 25.6 MB
    float* ssrc = (float*)(ws + 25600000);               // N*HEADS
    float* sdst = (float*)(ws + 26400000);               // N*HEADS
    float* nmax = (float*)(ws + 27200000);               // N*HEADS
    float* nsum = (float*)(ws + 28000000);               // N*HEADS

    gat_init     <<<(N_NODES * HC) / 256, 256, 0, stream>>>(out, bias, nmax, nsum);
    gat_gemm     <<<N_NODES / 16,          256, 0, stream>>>(x, W, h);
    gat_scores   <<<N_NODES / 8,           256, 0, stream>>>(h, att, ssrc, sdst);
    int eb = (N_ENTRIES + 255) / 256;
    gat_edge_max <<<eb,                    256, 0, stream>>>(ei, ssrc, sdst, nmax);
    gat_edge_sum <<<eb,                    256, 0, stream>>>(ei, ssrc, sdst, nmax, nsum);
    gat_aggregate<<<N_ENTRIES / 8,         256, 0, stream>>>(ei, ssrc, sdst, nmax, nsum, h, out);
}